// MVFeedforwardBlock_56057913147929
// MI455X (gfx1250) — compile-verified
//
#include <hip/hip_runtime.h>
#include <stdint.h>

#ifndef __has_builtin
#define __has_builtin(x) 0
#endif

typedef int v4i __attribute__((ext_vector_type(4)));
typedef __attribute__((address_space(1))) v4i gas_v4i;  // global (prints as __device__)
typedef __attribute__((address_space(3))) v4i las_v4i;  // LDS

// ===================== compile-time Clifford tables, PGA(3,0,1) =====================
// Blade order matches the reference: (),e0,e1,e2,e3,e01,e02,e03,e12,e13,e23,
// e012,e013,e023,e123,e0123.  metric: e0^2=0, e1..e3^2=1.
namespace ga {
constexpr int MASKS[16] = {0,1,2,4,8, 3,5,9,6,10,12, 7,11,13,14, 15};

constexpr int idxOfMask(int m) {
  for (int i = 0; i < 16; ++i) if (MASKS[i] == m) return i;
  return 0;
}
constexpr int gradeOf(int i) {
  int m = MASKS[i];
  return (m & 1) + ((m >> 1) & 1) + ((m >> 2) & 1) + ((m >> 3) & 1);
}
// #transpositions to interleave basis list a (ascending) with b (ascending):
// for each vector y in b, count vectors in a with index > y.
constexpr int swaps(int a, int b) {
  int s = 0;
  for (int y = 0; y < 4; ++y)
    if ((b >> y) & 1)
      for (int t = y + 1; t < 4; ++t)
        if ((a >> t) & 1) ++s;
  return s;
}
constexpr float sgn(int a, int b) { return (swaps(a, b) & 1) ? -1.0f : 1.0f; }

struct Tables {
  float gpS[16][16]; int gpI[16][16];  // geometric product: blade k is unique per (i,j)
  float jnS[16][16]; int jnI[16][16];  // join (regressive product), unique k per (i,j)
  float e0S[16];     int e0I[16];      // left-multiplication by e0 (sign is +1 or 0)
  int   grade[16];
};

constexpr Tables make() {
  Tables t{};
  for (int i = 0; i < 16; ++i) {
    t.grade[i] = gradeOf(i);
    int bm = MASKS[i];
    if (bm & 1) { t.e0S[i] = 0.0f; t.e0I[i] = 0; }                 // e0*e0 = 0
    else        { t.e0S[i] = 1.0f; t.e0I[i] = idxOfMask(bm | 1); } // sign always +1
  }
  for (int i = 0; i < 16; ++i)
    for (int j = 0; j < 16; ++j) {
      int a = MASKS[i], b = MASKS[j];
      // geometric product
      if (a & b & 1) { t.gpS[i][j] = 0.0f; t.gpI[i][j] = 0; }
      else           { t.gpS[i][j] = sgn(a, b); t.gpI[i][j] = idxOfMask(a ^ b); }
      // join: JOIN[k][i][j] = UND[k][c(k)] * OP[c(k)][c(i)][c(j)] * D[c(i)][i] * D[c(j)][j]
      // D is a signed permutation: D[c(i)][i] = 1/sgn(a, ~a); UND = D^-1.
      int ca = 15 ^ a, cb = 15 ^ b;
      if (ca & cb) { t.jnS[i][j] = 0.0f; t.jnI[i][j] = 0; }
      else {
        int k = idxOfMask(15 ^ (ca | cb));
        float d_i = 1.0f / sgn(a, 15 ^ a);
        float d_j = 1.0f / sgn(b, 15 ^ b);
        float u_k = sgn(MASKS[k], 15 ^ MASKS[k]);   // 1 / D[c(k)][k]
        t.jnS[i][j] = u_k * sgn(ca, cb) * d_i * d_j;
        t.jnI[i][j] = k;
      }
    }
  return t;
}
constexpr Tables T = make();
} // namespace ga

// out[k] += z[k]*w[grade(k)] + sum_j E0[k][j]*z[j]*v[grade(j)];  out[0] += bias
__device__ __forceinline__ void mv_linear_acc(const float z[16], const float* __restrict__ w,
                                              const float* __restrict__ v, float bias,
                                              float out[16]) {
#pragma unroll
  for (int k = 0; k < 16; ++k) out[k] += z[k] * w[ga::T.grade[k]];
#pragma unroll
  for (int j = 0; j < 16; ++j)
    if (ga::T.e0S[j] != 0.0f)                        // constant-folds after unroll
      out[ga::T.e0I[j]] += z[j] * v[ga::T.grade[j]]; // non-e0 blades have grade<=3
  out[0] += bias;
}

// ===================== pass 3 (defined first so disasm snippet shows it) =====================
// Streaming main kernel: 1 token/thread, async-staged 256-token tile in LDS.
__global__ __launch_bounds__(256) void mv_main(
    const float* __restrict__ x, const float* __restrict__ rnorm,
    const float* __restrict__ w1, const float* __restrict__ v1, const float* __restrict__ b1,
    const float* __restrict__ wg, const float* __restrict__ vg, const float* __restrict__ bg,
    const float* __restrict__ wj, const float* __restrict__ vj, const float* __restrict__ bj,
    float* __restrict__ out) {
  constexpr int TOK = 256;
  __shared__ __align__(16) float tile[TOK * 20];  // 80B token stride dodges bank conflicts
  const unsigned tid = threadIdx.x;
  const size_t gid = (size_t)blockIdx.x * TOK + tid;
  const int b = (int)(gid >> 15);  // 32768 tokens per batch, blocks never straddle batches
  const float* gsrc0 = x + ((size_t)blockIdx.x * TOK) * 16;

  // ---- CDNA5 async global->LDS staging of the 16KB tile (256 tokens x 64B) ----
#pragma unroll
  for (int c = 0; c < 4; ++c) {
    const float* g = gsrc0 + (size_t)tid * 16 + c * 4;
    float* l = &tile[tid * 20 + c * 4];
#if __has_builtin(__builtin_amdgcn_global_load_async_to_lds_b128)
    __builtin_amdgcn_global_load_async_to_lds_b128((gas_v4i*)g, (las_v4i*)l, 0, 0);
#else
    unsigned loff = (unsigned)(uintptr_t)(las_v4i*)l;
    asm volatile("global_load_async_to_lds_b128 %0, %1, off" ::"v"(loff), "v"(g) : "memory");
#endif
  }
#if __has_builtin(__builtin_amdgcn_s_wait_asynccnt)
  __builtin_amdgcn_s_wait_asynccnt(0);
#else
  asm volatile("s_wait_asynccnt 0" ::: "memory");
#endif
  __syncthreads();

  float xv[16];
#pragma unroll
  for (int c = 0; c < 4; ++c) {
    float4 t4 = *(const float4*)&tile[tid * 20 + c * 4];
    xv[4 * c + 0] = t4.x; xv[4 * c + 1] = t4.y; xv[4 * c + 2] = t4.z; xv[4 * c + 3] = t4.w;
  }

  // xn = x / norm
  const float rn = rnorm[b];
  float xn[16];
#pragma unroll
  for (int k = 0; k < 16; ++k) xn[k] = xv[k] * rn;

  // xp = mv_linear(xn; w1, v1, b1)
  float xp[16];
#pragma unroll
  for (int k = 0; k < 16; ++k) xp[k] = 0.0f;
  mv_linear_acc(xn, w1, v1, b1[0], xp);

  // gated = gelu(xp[0]) * xp   (exact gelu)
  const float x0 = xp[0];
  const float gel = 0.5f * x0 * (1.0f + erff(x0 * 0.70710678118654752f));
  float gated[16];
#pragma unroll
  for (int k = 0; k < 16; ++k) gated[k] = gel * xp[k];

  // gp[k] = sum GP[k][i][j] xn_i gated_j ;  jn[k] = sum JOIN[k][i][j] xn_i gated_j
  float gp[16], jn[16];
#pragma unroll
  for (int k = 0; k < 16; ++k) { gp[k] = 0.0f; jn[k] = 0.0f; }
#pragma unroll
  for (int i = 0; i < 16; ++i) {
    const float t = xn[i];
#pragma unroll
    for (int j = 0; j < 16; ++j) {
      if (ga::T.gpS[i][j] != 0.0f) gp[ga::T.gpI[i][j]] += ga::T.gpS[i][j] * (t * gated[j]);
      if (ga::T.jnS[i][j] != 0.0f) jn[ga::T.jnI[i][j]] += ga::T.jnS[i][j] * (t * gated[j]);
    }
  }
  const float xn15 = xn[15];
#pragma unroll
  for (int k = 0; k < 16; ++k) jn[k] *= xn15;

  // out = residual + mv_linear(gp; wg,vg,bg) + mv_linear(jn; wj,vj,bj)
  float o[16];
#pragma unroll
  for (int k = 0; k < 16; ++k) o[k] = xv[k];
  mv_linear_acc(gp, wg, vg, bg[0], o);
  mv_linear_acc(jn, wj, vj, bj[0], o);

  float4* op = (float4*)(out + gid * 16);
  op[0] = make_float4(o[0], o[1], o[2], o[3]);
  op[1] = make_float4(o[4], o[5], o[6], o[7]);
  op[2] = make_float4(o[8], o[9], o[10], o[11]);
  op[3] = make_float4(o[12], o[13], o[14], o[15]);
}

// ===================== pass 1: per-batch partial sums of sum(x*x*NON_E0) =====================
// grid: (16 blocks-per-batch, 16 batches), 256 threads, 2048 tokens per block
__global__ __launch_bounds__(256) void mv_reduce(const float* __restrict__ x,
                                                 float* __restrict__ partials) {
  const int b = blockIdx.y, blk = blockIdx.x;
  const float* xb = x + ((size_t)b * 32768 + (size_t)blk * 2048) * 16;
  float s = 0.0f;
#pragma unroll
  for (int it = 0; it < 8; ++it) {
    const int tok = it * 256 + threadIdx.x;
    const float4* p = (const float4*)(xb + (size_t)tok * 16);
    float4 a = p[0], c = p[1], d = p[2], e = p[3];
    // non-e0 blade indices: 0,2,3 | 4 | 8,9,10 | 14
    s += a.x * a.x + a.z * a.z + a.w * a.w;
    s += c.x * c.x;
    s += d.x * d.x + d.y * d.y + d.z * d.z;
    s += e.z * e.z;
  }
  __shared__ float red[256];
  red[threadIdx.x] = s;
  __syncthreads();
  for (int off = 128; off > 0; off >>= 1) {
    if (threadIdx.x < (unsigned)off) red[threadIdx.x] += red[threadIdx.x + off];
    __syncthreads();
  }
  if (threadIdx.x == 0) partials[b * 16 + blk] = red[0];
}

// ===================== pass 2: finalize 1/norm per batch (deterministic) =====================
__global__ void mv_norm(const float* __restrict__ partials, float* __restrict__ rnorm) {
  const int b = threadIdx.x;
  if (b < 16) {
    float s = 0.0f;
    for (int i = 0; i < 16; ++i) s += partials[b * 16 + i];
    rnorm[b] = 1.0f / sqrtf(s * (1.0f / 32768.0f) + 1e-5f);
  }
}

extern "C" void kernel_launch(void* const* d_in, const int* in_sizes, int n_in,
                              void* d_out, int out_size, void* d_ws, size_t ws_size,
                              hipStream_t stream) {
  (void)in_sizes; (void)n_in; (void)out_size; (void)ws_size;
  const float* x  = (const float*)d_in[0];
  const float* w1 = (const float*)d_in[1];
  const float* v1 = (const float*)d_in[2];
  const float* b1 = (const float*)d_in[3];
  const float* wg = (const float*)d_in[4];
  const float* vg = (const float*)d_in[5];
  const float* bg = (const float*)d_in[6];
  const float* wj = (const float*)d_in[7];
  const float* vj = (const float*)d_in[8];
  const float* bj = (const float*)d_in[9];
  float* out = (float*)d_out;

  float* partials = (float*)d_ws;      // 256 floats (written before read; no zeroing needed)
  float* rnorm    = partials + 256;    // 16 floats

  dim3 gridR(16, 16);
  mv_reduce<<<gridR, 256, 0, stream>>>(x, partials);
  mv_norm<<<1, 32, 0, stream>>>(partials, rnorm);
  mv_main<<<2048, 256, 0, stream>>>(x, rnorm, w1, v1, b1, wg, vg, bg, wj, vj, bj, out);
}